// GCN_86320252715431
// MI455X (gfx1250) — compile-verified
//
#include <hip/hip_runtime.h>
#include <math.h>

// ---------------------------------------------------------------------------
// Problem constants (from reference): B=64, O=64, F=1024, RN=16, RD=256, DEPTH=2
// ---------------------------------------------------------------------------
#define FDIM 1024
#define RDIM 256
#define RNUM 16
#define NROW 4096          // B*O flattened rows of x
#define NBATCH 64

typedef __attribute__((ext_vector_type(2))) float v2f;
typedef __attribute__((ext_vector_type(8))) float v8f;

// V_WMMA_F32_16X16X4_F32 : D(16x16 f32) = A(16x4 f32) * B(4x16 f32) + C
// 8-arg form: (neg_a, A, neg_b, B, c_mod, C, reuse_a, reuse_b)
__device__ __forceinline__ v8f wmma4(v2f a, v2f b, v8f c) {
  return __builtin_amdgcn_wmma_f32_16x16x4_f32(false, a, false, b, (short)0, c,
                                               false, false);
}

// ---------------------------------------------------------------------------
// Kernel 1: row reductions of A (layer invariant, run once).
//   AsumO[row, o'] = sum_r   A[row, o'*16 + r]   (row = b*64+o)
//   AsumR[row, r ] = sum_o'  A[row, o'*16 + r]
// ---------------------------------------------------------------------------
__global__ __launch_bounds__(64) void asum_kernel(const float* __restrict__ A,
                                                  float* __restrict__ AsumO,
                                                  float* __restrict__ AsumR) {
  const int row = blockIdx.x;       // 0..4095
  const int t = threadIdx.x;        // 0..63
  const float* a = A + (size_t)row * 1024;
  float s = 0.f;
#pragma unroll
  for (int r = 0; r < RNUM; ++r) s += a[t * RNUM + r];
  AsumO[row * 64 + t] = s;
  if (t < RNUM) {
    float sr = 0.f;
#pragma unroll 8
    for (int o = 0; o < 64; ++o) sr += a[o * RNUM + t];
    AsumR[row * RNUM + t] = sr;
  }
}

// ---------------------------------------------------------------------------
// Kernel 2: xr_r = R @ Wrel[i][F:, :] + brel[i]   -> [16, 1024]   (4 MFLOP)
// ---------------------------------------------------------------------------
__global__ __launch_bounds__(256) void xrr_kernel(const float* __restrict__ R,
                                                  const float* __restrict__ WrelR,
                                                  const float* __restrict__ brel,
                                                  float* __restrict__ xrr) {
  const int idx = blockIdx.x * 256 + threadIdx.x;   // 16384 total
  const int r = idx >> 10;
  const int n = idx & 1023;
  float s = brel[n];
#pragma unroll 8
  for (int k = 0; k < RDIM; ++k) s += R[r * RDIM + k] * WrelR[(size_t)k * FDIM + n];
  xrr[idx] = s;
}

// ---------------------------------------------------------------------------
// Kernel 3: dual-output fp32 WMMA GEMM sharing the A-operand (x):
//   xs  = X @ Wself + bself      [4096,1024]
//   xrx = X @ Wrel[:F]           [4096,1024]
// Block = 256 threads (8 waves), 64x64 output tile per matrix, BK=32.
// X tile staged m-major; W tiles staged TRANSPOSED (n-major) so every A and B
// fragment is one aligned ds_load_b64 (no VGPR repack moves).
// Row stride 34 floats = 136 B keeps 8-byte alignment and spreads lanes over
// distinct even bank pairs (17*n mod 32 is a permutation) -> conflict-free.
// ---------------------------------------------------------------------------
__global__ __launch_bounds__(256) void gemm_dual_kernel(
    const float* __restrict__ X, const float* __restrict__ Ws,
    const float* __restrict__ Wr, const float* __restrict__ bself,
    float* __restrict__ xs, float* __restrict__ xrx) {
  __shared__ float sX[64][34];        // X tile, [m][k]
  __shared__ float sWt[2][64][34];    // W tiles, [matrix][n][k] (transposed)

  const int tid = threadIdx.x;
  const int ln  = tid & 15;          // N (or M) index within fragment
  const int hi  = (tid >> 4) & 1;    // lane-half: K offset +2, M offset +8
  const int w   = tid >> 5;          // wave 0..7
  const int m0  = blockIdx.y * 64;
  const int n0  = blockIdx.x * 64;
  const int mt  = (w & 3) * 16;      // wave's m-tile inside block
  const int ntb = (w >> 2) * 32;     // wave's n-tile pair base

  v8f cs0 = {0, 0, 0, 0, 0, 0, 0, 0}, cs1 = {0, 0, 0, 0, 0, 0, 0, 0};
  v8f cr0 = {0, 0, 0, 0, 0, 0, 0, 0}, cr1 = {0, 0, 0, 0, 0, 0, 0, 0};

  for (int k0 = 0; k0 < FDIM; k0 += 32) {
    __syncthreads();
    {  // stage X tile: 64 rows x 32 cols, float4-coalesced global reads
      const int row = tid >> 3;
      const int c4 = (tid & 7) << 2;
      float4 v = *(const float4*)(X + (size_t)(m0 + row) * FDIM + k0 + c4);
      sX[row][c4 + 0] = v.x; sX[row][c4 + 1] = v.y;
      sX[row][c4 + 2] = v.z; sX[row][c4 + 3] = v.w;
      v = *(const float4*)(X + (size_t)(m0 + row + 32) * FDIM + k0 + c4);
      sX[row + 32][c4 + 0] = v.x; sX[row + 32][c4 + 1] = v.y;
      sX[row + 32][c4 + 2] = v.z; sX[row + 32][c4 + 3] = v.w;
    }
    {  // stage both W tiles 32x64, transposed into [n][k]
      const int row = tid >> 4;          // k-row 0..15 (+16)
      const int c4 = (tid & 15) << 2;    // n 0..60
#pragma unroll
      for (int rr = 0; rr < 2; ++rr) {
        const int kr = row + rr * 16;
        float4 v = *(const float4*)(Ws + (size_t)(k0 + kr) * FDIM + n0 + c4);
        sWt[0][c4 + 0][kr] = v.x; sWt[0][c4 + 1][kr] = v.y;
        sWt[0][c4 + 2][kr] = v.z; sWt[0][c4 + 3][kr] = v.w;
        v = *(const float4*)(Wr + (size_t)(k0 + kr) * FDIM + n0 + c4);
        sWt[1][c4 + 0][kr] = v.x; sWt[1][c4 + 1][kr] = v.y;
        sWt[1][c4 + 2][kr] = v.z; sWt[1][c4 + 3][kr] = v.w;
      }
    }
    __syncthreads();
#pragma unroll
    for (int kk = 0; kk < 32; kk += 4) {
      const int ka = kk + 2 * hi;   // even -> 8B-aligned v2f loads
      // A frag (16x4): lanes 0-15 carry K=kk..kk+1, lanes 16-31 K=kk+2..kk+3
      const v2f a = *(const v2f*)&sX[mt + ln][ka];
      // B frags (4x16): VGPR v holds K = kk + v + 2*hi, N = ln
      cs0 = wmma4(a, *(const v2f*)&sWt[0][ntb + ln][ka], cs0);
      cs1 = wmma4(a, *(const v2f*)&sWt[0][ntb + 16 + ln][ka], cs1);
      cr0 = wmma4(a, *(const v2f*)&sWt[1][ntb + ln][ka], cr0);
      cr1 = wmma4(a, *(const v2f*)&sWt[1][ntb + 16 + ln][ka], cr1);
    }
  }
  // C/D layout: VGPR v -> M = v + 8*hi, N = ln
#pragma unroll
  for (int v = 0; v < 8; ++v) {
    const int m = m0 + mt + v + 8 * hi;
    const int c0 = n0 + ntb + ln;
    const int c1 = c0 + 16;
    xs[(size_t)m * FDIM + c0] = cs0[v] + bself[c0];
    xs[(size_t)m * FDIM + c1] = cs1[v] + bself[c1];
    xrx[(size_t)m * FDIM + c0] = cr0[v];
    xrx[(size_t)m * FDIM + c1] = cr1[v];
  }
}

// ---------------------------------------------------------------------------
// Kernel 4: agg + residual + tanh (per batch b):
//   Out[b] = tanh( xs[b] + AsumO[b] @ xrx[b] + AsumR[b] @ xr_r + Xin[b] )
// One wave per 16x16 output tile; elementwise epilogue => in-place safe
// (Xin may alias Out).
// ---------------------------------------------------------------------------
__global__ __launch_bounds__(128) void agg_tanh_kernel(
    const float* __restrict__ AsumO, const float* __restrict__ AsumR,
    const float* __restrict__ xrx, const float* __restrict__ xrr,
    const float* __restrict__ xs, const float* __restrict__ Xin,
    float* __restrict__ Out) {
  const int tid = threadIdx.x;
  const int ln = tid & 15;
  const int hi = (tid >> 4) & 1;
  const int wv = tid >> 5;
  const int job = blockIdx.x * 4 + wv;   // 64 * 4 * 64 = 16384 tiles
  const int b  = job >> 8;
  const int ot = (job >> 6) & 3;
  const int nt = job & 63;
  const int col = nt * 16 + ln;
  const float* Ao = AsumO + (size_t)b * 64 * 64;
  const float* Ar = AsumR + (size_t)b * 64 * RNUM;

  v8f c = {0, 0, 0, 0, 0, 0, 0, 0};
#pragma unroll
  for (int k = 0; k < 64; k += 4) {   // AsumO[b] (64x64) @ xrx[b] (64x1024)
    const int ka = k + 2 * hi;
    // A rows are 64 floats (256 B): aligned v2f global loads
    const v2f a = *(const v2f*)(Ao + (ot * 16 + ln) * 64 + ka);
    v2f bb;
    bb[0] = xrx[(size_t)(b * 64 + ka) * FDIM + col];
    bb[1] = xrx[(size_t)(b * 64 + ka + 1) * FDIM + col];
    c = wmma4(a, bb, c);
  }
#pragma unroll
  for (int k = 0; k < RNUM; k += 4) {  // AsumR[b] (64x16) @ xr_r (16x1024)
    const int ka = k + 2 * hi;
    const v2f a = *(const v2f*)(Ar + (ot * 16 + ln) * RNUM + ka);
    v2f bb;
    bb[0] = xrr[(size_t)ka * FDIM + col];
    bb[1] = xrr[(size_t)(ka + 1) * FDIM + col];
    c = wmma4(a, bb, c);
  }
#pragma unroll
  for (int v = 0; v < 8; ++v) {
    const int m = b * 64 + ot * 16 + v + 8 * hi;
    const float val = c[v] + xs[(size_t)m * FDIM + col] + Xin[(size_t)m * FDIM + col];
    Out[(size_t)m * FDIM + col] = tanhf(val);
  }
}

// ---------------------------------------------------------------------------
// Host-side orchestration (graph-capture safe: only kernel launches on stream)
// ---------------------------------------------------------------------------
extern "C" void kernel_launch(void* const* d_in, const int* in_sizes, int n_in,
                              void* d_out, int out_size, void* d_ws, size_t ws_size,
                              hipStream_t stream) {
  const float* x     = (const float*)d_in[0];  // [64,64,1024]
  const float* A     = (const float*)d_in[1];  // [64,64,1024]
  const float* R     = (const float*)d_in[2];  // [16,256]
  const float* Wself = (const float*)d_in[3];  // [2,1024,1024]
  const float* bself = (const float*)d_in[4];  // [2,1024]
  const float* Wrel  = (const float*)d_in[5];  // [2,1280,1024]
  const float* brel  = (const float*)d_in[6];  // [2,1024]
  float* out = (float*)d_out;                  // [64,64,1024]

  // Workspace layout (floats): ~35 MB total
  float* ws   = (float*)d_ws;
  float* xs   = ws;                         // 4096*1024
  float* xrx  = xs + (size_t)NROW * FDIM;   // 4096*1024
  float* xrr  = xrx + (size_t)NROW * FDIM;  // 16*1024
  float* AsO  = xrr + RNUM * FDIM;          // 64*64*64
  float* AsR  = AsO + NBATCH * 64 * 64;     // 64*64*16

  // A-reductions are layer-invariant: compute once.
  asum_kernel<<<NROW, 64, 0, stream>>>(A, AsO, AsR);

  const float* Xl = x;
  for (int i = 0; i < 2; ++i) {
    const float* Ws_i  = Wself + (size_t)i * FDIM * FDIM;
    const float* WrF_i = Wrel + (size_t)i * (FDIM + RDIM) * FDIM;   // rows [0,F)
    const float* WrR_i = WrF_i + (size_t)FDIM * FDIM;               // rows [F,F+RD)

    xrr_kernel<<<64, 256, 0, stream>>>(R, WrR_i, brel + (size_t)i * FDIM, xrr);
    gemm_dual_kernel<<<dim3(16, 64), 256, 0, stream>>>(
        Xl, Ws_i, WrF_i, bself + (size_t)i * FDIM, xs, xrx);
    // Layer 0 writes d_out; layer 1 reads d_out elementwise and overwrites it
    // in place (safe: epilogue touches each element exactly once, read-then-write).
    agg_tanh_kernel<<<4096, 128, 0, stream>>>(AsO, AsR, xrx, xrr, xs, Xl, out);
    Xl = out;
  }
}